// ChamferLoss_24249385353750
// MI455X (gfx1250) — compile-verified
//
#include <hip/hip_runtime.h>

// Chamfer 1-NN loss on gfx1250 (MI455X).
//
// d(q,r) = |q|^2 + |r|^2 - 2 q.r ; min over refs ; sum ; * 2/B.
// Cross term as K=4 f32 WMMA: A(16 refs x 4) = [rx,ry,rz,|r|^2],
// B(4 x 16 queries) = [-2qx,-2qy,-2qz,1]  =>  D = |r|^2 - 2 q.r.
// Refs are staged in LDS as packed 16B records (x,y,z,r2) so the per-tile
// A-matrix fetch is a single ds_load_b64 matching the WMMA VGPR layout:
//   lanes 0-15 : (x,y)  = K0,K1      lanes 16-31 : (z,r2) = K2,K3
// Ref stream is double-buffered with async global->LDS (ASYNCcnt); the r2
// pack pass for chunk c+1 overlaps the WMMA processing of chunk c.

typedef float v2f __attribute__((ext_vector_type(2)));
typedef float v8f __attribute__((ext_vector_type(8)));

#define THREADS 256
#define CHUNK   256          // ref points staged per chunk
#define QT_PER_BLOCK 8       // one 16-query tile per wave, 8 waves/block

// Low 32 bits of a generic pointer to __shared__ == LDS byte offset.
__device__ __forceinline__ unsigned lds_off(const void* p) {
  return (unsigned)(unsigned long long)p;
}

__device__ __forceinline__ void async_ld_f32(const float* gsrc, unsigned ldst) {
  // GLOBAL_LOAD_ASYNC_TO_LDS_B32: vdst = LDS addr VGPR, vaddr = 64-bit global.
  asm volatile("global_load_async_to_lds_b32 %0, %1, off"
               :: "v"(ldst), "v"(gsrc) : "memory");
}

__device__ __forceinline__ void wait_async_le0() {
#if __has_builtin(__builtin_amdgcn_s_wait_asynccnt)
  __builtin_amdgcn_s_wait_asynccnt(0);
#else
  asm volatile("s_wait_asynccnt 0x0" ::: "memory");
#endif
}

__global__ __launch_bounds__(THREADS)
void chamfer_nn_kernel(const float* __restrict__ pc2,    // queries, [B,3,N] planes
                       const float* __restrict__ pc1w,   // refs,    [B,3,N] planes
                       float* __restrict__ partials, int N) {
  __shared__ float pk[2][CHUNK][4];     // double buffer of packed (x,y,z,r2)

  const int tid  = threadIdx.x;
  const int lane = tid & 31;
  const int wave = tid >> 5;

  const int qtilesPerBatch = N >> 4;
  const int blocksPerBatch = qtilesPerBatch / QT_PER_BLOCK;
  const int batch = blockIdx.x / blocksPerBatch;
  const int qtile = (blockIdx.x % blocksPerBatch) * QT_PER_BLOCK + wave;

  const float* q = pc2  + (size_t)batch * 3 * N;
  const float* r = pc1w + (size_t)batch * 3 * N;

  // ---- B matrix (queries), loaded once per wave -------------------------
  // 4x16 f32, 2 VGPRs: lanes 0-15 carry K=0/1, lanes 16-31 carry K=2/3.
  const int   n  = qtile * 16 + (lane & 15);
  const float qx = q[n];
  const float qy = q[N + n];
  const float qz = q[2 * N + n];
  const float q2 = qx * qx + qy * qy + qz * qz;
  v2f bm;
  bm.x = (lane < 16) ? (-2.0f * qx) : (-2.0f * qz);
  bm.y = (lane < 16) ? (-2.0f * qy) : 1.0f;

  // Per-lane base into the packed records: point (lane&15), half-select.
  const v2f* apBase0 = (const v2f*)(&pk[0][lane & 15][0] + ((lane < 16) ? 0 : 2));
  const v2f* apBase1 = (const v2f*)(&pk[1][lane & 15][0] + ((lane < 16) ? 0 : 2));

  v8f dmin;
#pragma unroll
  for (int i = 0; i < 8; ++i) dmin[i] = 3.0e38f;

  const int nChunks = N / CHUNK;

  // Prime: async chunk 0 -> buf 0, then pack r2 (own slot only).
  {
    async_ld_f32(r + tid,         lds_off(&pk[0][tid][0]));
    async_ld_f32(r + N + tid,     lds_off(&pk[0][tid][1]));
    async_ld_f32(r + 2 * N + tid, lds_off(&pk[0][tid][2]));
    wait_async_le0();
    const float x = pk[0][tid][0], y = pk[0][tid][1], z = pk[0][tid][2];
    pk[0][tid][3] = x * x + y * y + z * z;
  }

  for (int c = 0; c < nChunks; ++c) {
    const int buf = c & 1;
    const bool more = (c + 1 < nChunks);
    if (more) {                        // stream next chunk into other buffer
      const int m = (c + 1) * CHUNK + tid;
      async_ld_f32(r + m,         lds_off(&pk[buf ^ 1][tid][0]));
      async_ld_f32(r + N + m,     lds_off(&pk[buf ^ 1][tid][1]));
      async_ld_f32(r + 2 * N + m, lds_off(&pk[buf ^ 1][tid][2]));
    }
    __syncthreads();                   // all waves packed r2 for buf

    // ---- 16 ref tiles: ds_load_b64 -> WMMA -> 4x v_min3 ----------------
    const v2f* ap = buf ? apBase1 : apBase0;
#pragma unroll
    for (int t = 0; t < CHUNK / 16; ++t) {
      const v2f am = ap[t * 32];       // tile stride = 16 pts * 16B = 32 v2f
      v8f zero = {};
      v8f d = __builtin_amdgcn_wmma_f32_16x16x4_f32(
          false, am, false, bm, (short)0, zero, false, false);
#pragma unroll
      for (int i = 0; i < 8; ++i) dmin[i] = fminf(dmin[i], d[i]);
    }

    if (more) {                        // overlap: pack r2 for next chunk
      wait_async_le0();                // own async writes of chunk c+1 landed
      const int b2 = buf ^ 1;
      const float x = pk[b2][tid][0], y = pk[b2][tid][1], z = pk[b2][tid][2];
      pk[b2][tid][3] = x * x + y * y + z * z;
    }
    __syncthreads();                   // buf free for reuse next iteration
  }

  // ---- finalize: min over 8 VGPRs, then rows 0-7 vs 8-15 via SWAPX16 ----
  float pm = fminf(fminf(fminf(dmin[0], dmin[1]), fminf(dmin[2], dmin[3])),
                   fminf(fminf(dmin[4], dmin[5]), fminf(dmin[6], dmin[7])));
  pm = fminf(pm, __shfl_xor(pm, 16, 32));
  float res = fmaxf(pm + q2, 0.0f);   // clamp(min d, 0) == min clamp(d,0)

  // Sum the 16 query results of this wave (dup'd in both half-waves).
#pragma unroll
  for (int off = 8; off >= 1; off >>= 1) res += __shfl_xor(res, off, 32);
  if (lane == 0) partials[blockIdx.x * QT_PER_BLOCK + wave] = res;
}

__global__ __launch_bounds__(256)
void chamfer_reduce_kernel(const float* __restrict__ partials, float* __restrict__ out,
                           int n, float scale) {
  __shared__ float s[256];
  float sum = 0.0f;
  for (int i = threadIdx.x; i < n; i += 256) sum += partials[i];
  s[threadIdx.x] = sum;
  __syncthreads();
#pragma unroll
  for (int k = 128; k > 0; k >>= 1) {
    if (threadIdx.x < k) s[threadIdx.x] += s[threadIdx.x + k];
    __syncthreads();
  }
  if (threadIdx.x == 0) out[0] = s[0] * scale;
}

extern "C" void kernel_launch(void* const* d_in, const int* in_sizes, int n_in,
                              void* d_out, int out_size, void* d_ws, size_t ws_size,
                              hipStream_t stream) {
  const float* pc2  = (const float*)d_in[0];   // queries [B,3,N]
  const float* pc1w = (const float*)d_in[1];   // refs    [B,3,N]
  const int B = 4, C = 3;
  const int N = in_sizes[0] / (B * C);         // 8192

  float* partials = (float*)d_ws;              // blocks*8 floats (8 KB)

  const int blocksPerBatch = (N / 16) / QT_PER_BLOCK;
  const int blocks = B * blocksPerBatch;       // 256

  chamfer_nn_kernel<<<blocks, THREADS, 0, stream>>>(pc2, pc1w, partials, N);
  chamfer_reduce_kernel<<<1, 256, 0, stream>>>(partials, (float*)d_out,
                                               blocks * QT_PER_BLOCK, 2.0f / (float)B);
}